// LSSTransformStandalone_2860448219649
// MI455X (gfx1250) — compile-verified
//
#include <hip/hip_runtime.h>
#include <hip/hip_bf16.h>

// ---------------- problem constants (match reference) ----------------
#define IN_CH   256
#define OUT_CH  80
#define DBINS   68          // (35 - 1) / 0.5
#define ZCH     148         // DBINS + OUT_CH
#define NCAM    6
#define BSZ     2
#define BN      12          // BSZ * NCAM
#define FH      30
#define FW      50
#define HW      1500        // FH * FW
#define NX0     100
#define NX1     200

// padded z tile store: 160 rows x 1504 cols per camera (garbage in pads, never read)
#define ZROWS   160
#define ZCOLS   1504
#define ZPITCH  (ZROWS * ZCOLS)              // floats per bn
#define Z_ELEMS (BN * ZPITCH)                // 2,887,680 floats in ws
#define BEV_ELEMS (BSZ * OUT_CH * NX1 * NX0) // 3,200,000 floats (d_out head)

#define MTILES_PER_WAVE 5                    // 10 M-tiles total -> 2 groups

typedef float v2f __attribute__((ext_vector_type(2)));
typedef float v8f __attribute__((ext_vector_type(8)));

// ---------------- small 3x3 helpers ----------------
__device__ inline void inv3(const float* m, float* o) {
    float a = m[0], b = m[1], c = m[2];
    float d = m[3], e = m[4], f = m[5];
    float g = m[6], h = m[7], i = m[8];
    float A =  (e * i - f * h);
    float B = -(d * i - f * g);
    float C =  (d * h - e * g);
    float det = a * A + b * B + c * C;
    float inv = 1.0f / det;
    o[0] = A * inv;              o[1] = -(b * i - c * h) * inv; o[2] =  (b * f - c * e) * inv;
    o[3] = B * inv;              o[4] =  (a * i - c * g) * inv; o[5] = -(a * f - c * d) * inv;
    o[6] = C * inv;              o[7] = -(a * h - b * g) * inv; o[8] =  (a * e - b * d) * inv;
}

__device__ inline void mat3mul(const float* x, const float* y, float* o) {
    for (int i = 0; i < 3; ++i)
        for (int j = 0; j < 3; ++j)
            o[i * 3 + j] = x[i * 3 + 0] * y[0 * 3 + j]
                         + x[i * 3 + 1] * y[1 * 3 + j]
                         + x[i * 3 + 2] * y[2 * 3 + j];
}

// ---------------- kernel 1: fold per-camera transforms ----------------
// per bn: [0..8]=inv(post_rots), [9..11]=post_trans,
//         [12..20]=inv(l2e_R)*rots*inv(K), [21..23]=inv(l2e_R)*(trans - l2e_t)
__global__ void prep_kernel(const float* __restrict__ c2e,
                            const float* __restrict__ intr,
                            const float* __restrict__ l2e,
                            const float* __restrict__ aug,
                            float* __restrict__ mats) {
    int t = threadIdx.x;
    if (t >= BN) return;
    int b = t / NCAM;

    const float* A4 = aug + t * 16;
    float P[9], Pt[3];
    for (int i = 0; i < 3; ++i) {
        for (int j = 0; j < 3; ++j) P[i * 3 + j] = A4[i * 4 + j];
        Pt[i] = A4[i * 4 + 3];
    }
    float invP[9]; inv3(P, invP);

    const float* Km = intr + t * 9;
    float K[9];
    for (int i = 0; i < 9; ++i) K[i] = Km[i];
    float invK[9]; inv3(K, invK);

    const float* C4 = c2e + t * 16;
    float R[9], tr[3];
    for (int i = 0; i < 3; ++i) {
        for (int j = 0; j < 3; ++j) R[i * 3 + j] = C4[i * 4 + j];
        tr[i] = C4[i * 4 + 3];
    }
    float comb[9]; mat3mul(R, invK, comb);

    const float* L4 = l2e + b * 16;
    float L[9], lt[3];
    for (int i = 0; i < 3; ++i) {
        for (int j = 0; j < 3; ++j) L[i * 3 + j] = L4[i * 4 + j];
        lt[i] = L4[i * 4 + 3];
    }
    float invL[9]; inv3(L, invL);

    float Af[9]; mat3mul(invL, comb, Af);
    float dt[3] = { tr[0] - lt[0], tr[1] - lt[1], tr[2] - lt[2] };
    float tv[3];
    for (int i = 0; i < 3; ++i)
        tv[i] = invL[i * 3 + 0] * dt[0] + invL[i * 3 + 1] * dt[1] + invL[i * 3 + 2] * dt[2];

    float* o = mats + t * 24;
    for (int i = 0; i < 9; ++i) o[i]      = invP[i];
    for (int i = 0; i < 3; ++i) o[9 + i]  = Pt[i];
    for (int i = 0; i < 9; ++i) o[12 + i] = Af[i];
    for (int i = 0; i < 3; ++i) o[21 + i] = tv[i];
}

// ---------------- kernel 2: z = W @ img + bias via V_WMMA_F32_16X16X4_F32 ----------------
// One wave: 5 M-tiles (80 rows) x 16 pixels, K=256 in 64 wmma steps per tile.
// B fragment loaded once per K-step and reused across the 5 M-tiles.
// No masking: out-of-range A rows / B cols use clamped addresses and their
// garbage products land in padded z rows/cols that are never read.
__global__ __launch_bounds__(32) void gemm_wmma_kernel(const float* __restrict__ img,
                                                       const float* __restrict__ W,
                                                       const float* __restrict__ bias,
                                                       float* __restrict__ z) {
    const int lane = threadIdx.x;
    const int lm   = lane & 15;
    const int half = lane >> 4;              // 0: K pair {k,k+1}, 1: {k+2,k+3}
    const int hw0  = blockIdx.x * 16;        // pixel tile base (0..1488)
    const int mg   = blockIdx.y;             // M-tile group (0 or 1)
    const int bn   = blockIdx.z;

    // A row pointers per M-tile (clamped; pad rows read row 147's data)
    const float* wp[MTILES_PER_WAVE];
    #pragma unroll
    for (int t = 0; t < MTILES_PER_WAVE; ++t) {
        int row = (mg * MTILES_PER_WAVE + t) * 16 + lm;
        wp[t] = W + (size_t)((row < ZCH) ? row : (ZCH - 1)) * IN_CH;
    }

    // B column pointer (clamped; pad cols read col 1499's data)
    const int col = hw0 + lm;
    const float* ip = img + (size_t)bn * IN_CH * HW + ((col < HW) ? col : (HW - 1));

    v8f acc[MTILES_PER_WAVE];
    #pragma unroll
    for (int t = 0; t < MTILES_PER_WAVE; ++t) acc[t] = (v8f){};

    for (int k = 0; k < IN_CH; k += 4) {
        const int ka = k + 2 * half;
        v2f bfr;
        bfr.x = ip[(size_t)ka * HW];         // img[ka][col]
        bfr.y = ip[(size_t)(ka + 1) * HW];   // img[ka+1][col]
        #pragma unroll
        for (int t = 0; t < MTILES_PER_WAVE; ++t) {
            const v2f a = *(const v2f*)(wp[t] + ka);   // W[row][ka], W[row][ka+1]
            acc[t] = __builtin_amdgcn_wmma_f32_16x16x4_f32(
                false, a, false, bfr, (short)0, acc[t], false, false);
        }
    }

    // D layout: acc[t][r] = (M = mt*16 + r + 8*half, N = hw0 + lm)
    // unconditional stores into the padded z surface
    float* zb = z + (size_t)bn * ZPITCH + (size_t)col; // note: col is the real column index
    float* zcol = z + (size_t)bn * ZPITCH + (size_t)(hw0 + lm);
    (void)zb;
    #pragma unroll
    for (int t = 0; t < MTILES_PER_WAVE; ++t) {
        const int mbase = (mg * MTILES_PER_WAVE + t) * 16 + 8 * half;
        #pragma unroll
        for (int r = 0; r < 8; ++r) {
            const int row = mbase + r;
            const float bv = bias[(row < ZCH) ? row : (ZCH - 1)];
            zcol[(size_t)row * ZCOLS] = acc[t][r] + bv;
        }
    }
}

// ---------------- kernel 3: softmax over first DBINS channels ----------------
__global__ void softmax_kernel(const float* __restrict__ z, float* __restrict__ depth) {
    const int idx = blockIdx.x * blockDim.x + threadIdx.x;
    if (idx >= BN * HW) return;
    const int bn = idx / HW;
    const int hw = idx % HW;
    const float* zp = z + (size_t)bn * ZPITCH + hw;
    float m = -3.0e38f;
    for (int d = 0; d < DBINS; ++d) m = fmaxf(m, zp[(size_t)d * ZCOLS]);
    float* dp = depth + (size_t)bn * DBINS * HW + hw;
    float s = 0.0f;
    for (int d = 0; d < DBINS; ++d) {
        float e = expf(zp[(size_t)d * ZCOLS] - m);
        s += e;
        dp[(size_t)d * HW] = e;
    }
    const float inv = 1.0f / s;
    for (int d = 0; d < DBINS; ++d) dp[(size_t)d * HW] *= inv;
}

// ---------------- kernel 4: geometry + voxel splat (atomic scatter-add) ----------------
__global__ void splat_kernel(const float* __restrict__ z,
                             const float* __restrict__ mats,
                             const float* __restrict__ depth,
                             float* __restrict__ bev) {
    const int idx = blockIdx.x * blockDim.x + threadIdx.x;
    if (idx >= BN * DBINS * HW) return;
    const int bn  = idx / (DBINS * HW);
    const int rem = idx % (DBINS * HW);
    const int d   = rem / HW;
    const int hw  = rem % HW;
    const int h   = hw / FW;
    const int w   = hw % FW;

    const float* M = mats + bn * 24;

    // frustum point (image x, image y, depth), iW=800 iH=480
    const float xs = (float)w * (799.0f / 49.0f);
    const float ys = (float)h * (479.0f / 29.0f);
    const float ds = 1.0f + 0.5f * (float)d;

    // undo post-aug
    const float p0 = xs - M[9], p1 = ys - M[10], p2 = ds - M[11];
    const float q0 = M[0] * p0 + M[1] * p1 + M[2] * p2;
    const float q1 = M[3] * p0 + M[4] * p1 + M[5] * p2;
    const float q2 = M[6] * p0 + M[7] * p1 + M[8] * p2;
    // unproject
    const float s0 = q0 * q2, s1 = q1 * q2, s2 = q2;
    // to lidar frame
    const float g0 = M[12] * s0 + M[13] * s1 + M[14] * s2 + M[21];
    const float g1 = M[15] * s0 + M[16] * s1 + M[17] * s2 + M[22];
    const float g2 = M[18] * s0 + M[19] * s1 + M[20] * s2 + M[23];

    // voxelize; offsets computed exactly as reference (BX - DX/2) in fp32
    const float offx = (-15.0f + 0.15f) - 0.15f;
    const float offy = (-30.0f + 0.15f) - 0.15f;
    const float offz = (-2.0f + 2.0f) - 2.0f;
    const int gx = (int)((g0 - offx) / 0.3f);   // truncation toward zero == astype(int32)
    const int gy = (int)((g1 - offy) / 0.3f);
    const int gz = (int)((g2 - offz) / 4.0f);
    if (gx < 0 || gx >= NX0 || gy < 0 || gy >= NX1 || gz != 0) return;

    const float dv = depth[(size_t)(bn * DBINS + d) * HW + hw];
    const int b = bn / NCAM;
    const float* feat = z + (size_t)bn * ZPITCH + (size_t)DBINS * ZCOLS + hw;
    // bev[b, c, gy, gx] with shape (B, OUT_CH, NX1, NX0)
    float* ob = bev + (((size_t)b * OUT_CH) * NX1 + gy) * NX0 + gx;
    #pragma unroll 4
    for (int c = 0; c < OUT_CH; ++c) {
        atomicAdd(ob + (size_t)c * NX1 * NX0, dv * feat[(size_t)c * ZCOLS]);
    }
}

// ---------------- launch ----------------
extern "C" void kernel_launch(void* const* d_in, const int* in_sizes, int n_in,
                              void* d_out, int out_size, void* d_ws, size_t ws_size,
                              hipStream_t stream) {
    const float* images = (const float*)d_in[0]; // (12, 256, 30, 50)
    const float* c2e    = (const float*)d_in[1]; // (2, 6, 4, 4)
    const float* intr   = (const float*)d_in[2]; // (2, 6, 3, 3)
    const float* l2e    = (const float*)d_in[3]; // (2, 4, 4)
    const float* aug    = (const float*)d_in[4]; // (2, 6, 4, 4)
    const float* W      = (const float*)d_in[5]; // (148, 256)
    const float* bias   = (const float*)d_in[6]; // (148,)

    float* out  = (float*)d_out;
    float* ws   = (float*)d_ws;
    float* z    = ws;                 // 12 * 160 * 1504 floats (padded)
    float* mats = ws + Z_ELEMS;       // 12 * 24 floats
    float* bev   = out;               // (2, 80, 200, 100)
    float* depth = out + BEV_ELEMS;   // (2, 6, 68, 30, 50)

    hipMemsetAsync(bev, 0, (size_t)BEV_ELEMS * sizeof(float), stream);

    prep_kernel<<<1, 32, 0, stream>>>(c2e, intr, l2e, aug, mats);

    dim3 ggrid((ZCOLS) / 16, ZROWS / (16 * MTILES_PER_WAVE), BN);   // 94 x 2 x 12 waves
    gemm_wmma_kernel<<<ggrid, 32, 0, stream>>>(images, W, bias, z);

    const int npix = BN * HW;
    softmax_kernel<<<(npix + 255) / 256, 256, 0, stream>>>(z, depth);

    const int npts = BN * DBINS * HW;
    splat_kernel<<<(npts + 255) / 256, 256, 0, stream>>>(z, mats, depth, bev);
}